// GraphSAGEv2_12704513261865
// MI455X (gfx1250) — compile-verified
//
#include <hip/hip_runtime.h>

typedef __attribute__((ext_vector_type(2))) float v2f;
typedef __attribute__((ext_vector_type(8))) float v8f;

#define CCH 128          // feature channels
#define LSTR 132         // LDS row stride in floats (conflict-free b64 reads)

// ---------------- utility kernels ----------------

__global__ void zero_f32(float* __restrict__ p, long long n) {
  long long i = (long long)blockIdx.x * blockDim.x + threadIdx.x;
  long long stride = (long long)gridDim.x * blockDim.x;
  for (; i < n; i += stride) p[i] = 0.0f;
}

__global__ void deg_count(const long long* __restrict__ dst, float* __restrict__ deg, int E) {
  int e = blockIdx.x * blockDim.x + threadIdx.x;
  if (e < E) atomicAdd(&deg[(int)dst[e]], 1.0f);
}

// 2 edges per 256-thread block; 128 channels per edge, coalesced atomics.
__global__ void scatter_add(const long long* __restrict__ ei,  // [2][E]
                            const float* __restrict__ x,       // [N][C]
                            float* __restrict__ agg,           // [N][C]
                            int E) {
  int ledge = threadIdx.x >> 7;   // 0..1
  int c = threadIdx.x & 127;
  int e = blockIdx.x * 2 + ledge;
  if (e >= E) return;
  int s = (int)ei[e];
  int d = (int)ei[(long long)E + e];
  atomicAdd(&agg[(long long)d * CCH + c], x[(long long)s * CCH + c]);
}

// ---------------- fused SAGEConv + LayerNorm + ReLU ----------------
// One block (8 wave32s) per 16-row node tile.
// h = (agg*inv_deg) @ Wl^T + bl + x @ Wr^T ; LN over C ; ReLU
__global__ __launch_bounds__(256, 2)
void sage_layer(const float* __restrict__ xin,   // [N][C]
                const float* __restrict__ agg,   // [N][C] (unscaled sums)
                const float* __restrict__ deg,   // [N]
                const float* __restrict__ Wl,    // [C][C] row-major (out,in)
                const float* __restrict__ bl,    // [C]
                const float* __restrict__ Wr,    // [C][C]
                const float* __restrict__ lnw,   // [C]
                const float* __restrict__ lnb,   // [C]
                float* __restrict__ xout,        // [N][C]
                int N) {
  __shared__ float sA[16 * LSTR];  // inv_deg-scaled aggregate tile
  __shared__ float sX[16 * LSTR];  // x tile
  __shared__ float sH[16 * LSTR];  // pre-LN result tile

  const int tid = threadIdx.x;
  const int rowBase = blockIdx.x * 16;

  // Stage 16x128 tiles (8 elements / thread), fold mean-scaling into agg.
  for (int i = tid; i < 16 * CCH; i += 256) {
    int r = i >> 7;
    int c = i & 127;
    int grow = rowBase + r;
    float vx = 0.0f, va = 0.0f;
    if (grow < N) {
      vx = xin[(long long)grow * CCH + c];
      float d = deg[grow];
      float sc = (d > 0.0f) ? (1.0f / d) : 0.0f;
      va = agg[(long long)grow * CCH + c] * sc;
    }
    sX[r * LSTR + c] = vx;
    sA[r * LSTR + c] = va;
  }
  __syncthreads();

  const int wave = tid >> 5;       // 0..7 -> output column tile
  const int lane = tid & 31;
  const int half = lane >> 4;      // 0/1
  const int col  = lane & 15;      // M for A-frag, N-col for B/C/D frags
  const int nBase = wave * 16;

  // A fragment: a[0]=A[M=col][K=2*half+k0], a[1]=A[col][2*half+1+k0] (ISA 7.12.2)
  const float* apA = sA + col * LSTR + 2 * half;
  const float* apX = sX + col * LSTR + 2 * half;
  // B fragment: B(k,n)=W[n][k]; n=nBase+col; consecutive k -> b64 load
  const float* wlp = Wl + (long long)(nBase + col) * CCH + 2 * half;
  const float* wrp = Wr + (long long)(nBase + col) * CCH + 2 * half;

  // Two independent accumulator chains (agg@Wl^T and x@Wr^T) so the matrix
  // pipe sees two interleavable WMMA dependency chains instead of one.
  v8f accL = {};
  v8f accR = {};
  #pragma unroll 4
  for (int k0 = 0; k0 < CCH; k0 += 4) {
    v2f aA = *(const v2f*)(apA + k0);
    v2f aX = *(const v2f*)(apX + k0);
    v2f bL = *(const v2f*)(wlp + k0);
    v2f bR = *(const v2f*)(wrp + k0);
    accL = __builtin_amdgcn_wmma_f32_16x16x4_f32(false, aA, false, bL,
                                                 (short)0, accL, false, false);
    accR = __builtin_amdgcn_wmma_f32_16x16x4_f32(false, aX, false, bR,
                                                 (short)0, accR, false, false);
  }

  // bias depends only on output column
  float bias = bl[nBase + col];
  #pragma unroll
  for (int j = 0; j < 8; ++j) {
    sH[(j + 8 * half) * LSTR + nBase + col] = accL[j] + accR[j] + bias;
  }
  __syncthreads();

  // LayerNorm + ReLU: 16 threads per row, 8 columns each.
  const int r  = tid >> 4;
  const int ci = tid & 15;
  float vals[8];
  float s = 0.0f;
  #pragma unroll
  for (int j = 0; j < 8; ++j) {
    float v = sH[r * LSTR + ci + 16 * j];
    vals[j] = v;
    s += v;
  }
  #pragma unroll
  for (int off = 8; off >= 1; off >>= 1) s += __shfl_xor(s, off, 32);
  float mu = s * (1.0f / CCH);
  float s2 = 0.0f;
  #pragma unroll
  for (int j = 0; j < 8; ++j) { float d = vals[j] - mu; s2 += d * d; }
  #pragma unroll
  for (int off = 8; off >= 1; off >>= 1) s2 += __shfl_xor(s2, off, 32);
  float rstd = rsqrtf(s2 * (1.0f / CCH) + 1e-5f);

  int grow = rowBase + r;
  if (grow < N) {
    #pragma unroll
    for (int j = 0; j < 8; ++j) {
      int c = ci + 16 * j;
      float v = (vals[j] - mu) * rstd * lnw[c] + lnb[c];
      xout[(long long)grow * CCH + c] = fmaxf(v, 0.0f);
    }
  }
}

// ---------------- launcher ----------------

extern "C" void kernel_launch(void* const* d_in, const int* in_sizes, int n_in,
                              void* d_out, int out_size, void* d_ws, size_t ws_size,
                              hipStream_t stream) {
  const float*     x   = (const float*)d_in[0];
  const long long* ei  = (const long long*)d_in[1];
  const float*     Wl  = (const float*)d_in[2];
  const float*     bl  = (const float*)d_in[3];
  const float*     Wr  = (const float*)d_in[4];
  const float*     lnw = (const float*)d_in[5];
  const float*     lnb = (const float*)d_in[6];

  const int N = in_sizes[0] / CCH;   // 100000
  const int E = in_sizes[1] / 2;     // 1600000
  const int L = in_sizes[3] / CCH;   // 4

  float* ws   = (float*)d_ws;
  float* deg  = ws;                              // N floats
  float* agg  = ws + N;                          // N*C floats
  float* bufA = agg + (size_t)N * CCH;           // N*C floats

  // in-degree (computed once; inversion folded into sage_layer staging)
  zero_f32<<<1024, 256, 0, stream>>>(deg, (long long)N);
  deg_count<<<(E + 255) / 256, 256, 0, stream>>>(ei + E, deg, E);

  const int nblocks = (N + 15) / 16;             // 6250, exact for N=100000
  const float* cur = x;
  for (int l = 0; l < L; ++l) {
    zero_f32<<<4096, 256, 0, stream>>>(agg, (long long)N * CCH);
    scatter_add<<<(E + 1) / 2, 256, 0, stream>>>(ei, cur, agg, E);
    // ping-pong: even layers -> bufA, odd -> d_out; L=4 so layer 3 ends in d_out
    float* outbuf = (l == L - 1) ? (float*)d_out
                                 : ((l & 1) == 0 ? bufA : (float*)d_out);
    sage_layer<<<nblocks, 256, 0, stream>>>(
        cur, agg, deg,
        Wl + (size_t)l * CCH * CCH, bl + (size_t)l * CCH,
        Wr + (size_t)l * CCH * CCH,
        lnw + (size_t)l * CCH, lnb + (size_t)l * CCH,
        outbuf, N);
    cur = outbuf;
  }
}